// Routing_1133871366146
// MI455X (gfx1250) — compile-verified
//
#include <hip/hip_runtime.h>

// Capsule dynamic routing, fully fused for gfx1250 (MI455X).
// - u is never materialized (saves ~2.3 GB of HBM traffic); it is recomputed
//   from x and W (both L2-resident) via V_WMMA_F32_16X16X4_F32.
// - Routing logits are linear in the accumulated v, so only Vacc (10KB LDS)
//   is carried between iterations -> whole routing loop fuses in one kernel.
// - Fragment orientation: C[M=j][N=batch] so softmax is per-lane and all
//   j-reductions are a single shfl_xor(16) (lane L / L^16 share a batch).
// - launch_bounds(TPB, 1): let the RA use >256 VGPRs (gfx1250 VGPR-MSB
//   addressing) so s_frag[10]+u_frag[10] (160 VGPRs) live without spilling.

typedef __attribute__((ext_vector_type(2))) float v2f;
typedef __attribute__((ext_vector_type(8))) float v8f;

#define NI 1152
#define NO 10
#define DA 16
#define DK 8
#define WAVES 16
#define TPB (WAVES * 32)

static __device__ inline v8f v8f_zero() {
  v8f z = {0.f, 0.f, 0.f, 0.f, 0.f, 0.f, 0.f, 0.f};
  return z;
}

__global__ __launch_bounds__(TPB, 1) void caps_route_wmma(
    const float* __restrict__ x,        // [B, NI, DK]
    const float* __restrict__ W,        // [NI, NO, DA, DK]
    const int* __restrict__ n_iter_p,   // scalar
    float* __restrict__ out,            // [B, NO, DA]
    int B)
{
  __shared__ float s_sum[NO * DA * 16]; // [o][j][batch]  10 KB
  __shared__ float vacc [NO * 16 * DA]; // [o][batch][j]  10 KB

  const int n_iter = *n_iter_p;
  const int tid  = threadIdx.x;
  const int lane = tid & 31;
  const int wave = tid >> 5;
  const int b0   = blockIdx.x * 16;

  const int bl = lane & 15;   // batch-in-tile (N) / row-in-tile (M)
  const int hi = lane >> 4;   // lane half: selects K pair / j half
  const int k0 = hi * 2;      // WMMA f32 16x16x4: lanes 0-15 hold K=0,1; 16-31 hold K=2,3

  for (int t = tid; t < NO * 16 * DA; t += TPB) vacc[t] = 0.0f;
  __syncthreads();

  const int brow = (b0 + bl < B) ? (b0 + bl) : (B - 1);
  const float* xrow = x + (size_t)brow * NI * DK;

  v8f s_frag[NO];   // partial s[o][j(frag)][batch(lane)] for this wave's i's
  v8f u_frag[NO];   // u tile for current i, all 10 capsules

  for (int it = 0; it < n_iter; ++it) {
    #pragma unroll
    for (int o = 0; o < NO; ++o) s_frag[o] = v8f_zero();

    for (int i = wave; i < NI; i += WAVES) {
      // B fragment (4x16, K x batch) from x[b, i, :]
      v2f b_lo = *(const v2f*)(xrow + i * DK + k0);       // K = k0, k0+1
      v2f b_hi = *(const v2f*)(xrow + i * DK + k0 + 4);   // K = k0+4, k0+5
      const float* wbase = W + (size_t)i * (NO * DA * DK);

      float l[NO];
      #pragma unroll
      for (int o = 0; o < NO; ++o) {
        // A fragment (16x4, j x K) from W[i, o, :, :]
        const float* wrow = wbase + (o * DA + bl) * DK;
        v2f a_lo = *(const v2f*)(wrow + k0);
        v2f a_hi = *(const v2f*)(wrow + k0 + 4);

        v8f acc = v8f_zero();
        acc = __builtin_amdgcn_wmma_f32_16x16x4_f32(false, a_lo, false, b_lo,
                                                    (short)0, acc, false, false);
        acc = __builtin_amdgcn_wmma_f32_16x16x4_f32(false, a_hi, false, b_hi,
                                                    (short)0, acc, false, false);
        u_frag[o] = acc; // C[M=j][N=batch]: VGPR r -> j=r+8*hi, lane -> batch=bl

        // logit partial: sum_j u * Vacc (this lane's 8 j's)
        const float* vp = &vacc[(o * 16 + bl) * DA + hi * 8];
        float t = 0.f;
        #pragma unroll
        for (int r = 0; r < 8; ++r) t = fmaf(acc[r], vp[r], t);
        l[o] = t;
      }
      // complete j-reduction: lane and lane^16 hold complementary j halves
      #pragma unroll
      for (int o = 0; o < NO; ++o) l[o] += __shfl_xor(l[o], 16);

      // per-lane softmax over the 10 capsules (iter 0: logits==0 -> uniform)
      float mx = l[0];
      #pragma unroll
      for (int o = 1; o < NO; ++o) mx = fmaxf(mx, l[o]);
      float c[NO];
      float sum = 0.f;
      #pragma unroll
      for (int o = 0; o < NO; ++o) { c[o] = __expf(l[o] - mx); sum += c[o]; }
      float inv = __builtin_amdgcn_rcpf(sum);  // v_rcp_f32: softmax normalize
      #pragma unroll
      for (int o = 0; o < NO; ++o) {
        float cv = c[o] * inv;
        #pragma unroll
        for (int r = 0; r < 8; ++r)
          s_frag[o][r] = fmaf(cv, u_frag[o][r], s_frag[o][r]);
      }
    }

    // cross-wave reduction of s into LDS
    for (int t = tid; t < NO * DA * 16; t += TPB) s_sum[t] = 0.f;
    __syncthreads();
    #pragma unroll
    for (int o = 0; o < NO; ++o) {
      #pragma unroll
      for (int r = 0; r < 8; ++r)
        atomicAdd(&s_sum[(o * DA + hi * 8 + r) * 16 + bl], s_frag[o][r]);
    }
    __syncthreads();

    // squash + Vacc update (+ final output) by wave 0 only
    if (wave == 0) {
      #pragma unroll
      for (int o = 0; o < NO; ++o) {
        float sv[8];
        float nrm = 0.f;
        #pragma unroll
        for (int r = 0; r < 8; ++r) {
          sv[r] = s_sum[(o * DA + hi * 8 + r) * 16 + bl];
          nrm = fmaf(sv[r], sv[r], nrm);
        }
        nrm += __shfl_xor(nrm, 16);                     // full |s|^2 over 16 j
        float nn = sqrtf(nrm);
        float scale = nrm / ((1.0f + nrm) * fmaxf(nn, 1e-30f));
        #pragma unroll
        for (int r = 0; r < 8; ++r) {
          float vv = scale * sv[r];
          vacc[(o * 16 + bl) * DA + hi * 8 + r] += vv;  // logits linear in Vacc
          if (it == n_iter - 1 && b0 + bl < B)
            out[((size_t)(b0 + bl) * NO + o) * DA + hi * 8 + r] = vv;
        }
      }
    }
    __syncthreads();
  }
}

extern "C" void kernel_launch(void* const* d_in, const int* in_sizes, int n_in,
                              void* d_out, int out_size, void* d_ws, size_t ws_size,
                              hipStream_t stream) {
  (void)n_in; (void)out_size; (void)d_ws; (void)ws_size;
  const float* x      = (const float*)d_in[0];
  const float* W      = (const float*)d_in[1];
  const int*   n_iter = (const int*)d_in[2];
  float* out = (float*)d_out;

  int B = in_sizes[0] / (NI * DK);
  int blocks = (B + 15) / 16;
  caps_route_wmma<<<blocks, TPB, 0, stream>>>(x, W, n_iter, out, B);
}